// PCSCB_1305670058631
// MI455X (gfx1250) — compile-verified
//
#include <hip/hip_runtime.h>
#include <hip/hip_bf16.h>
#include <math.h>

// Problem constants (match reference)
#define BB   64
#define CC   128
#define LL   4000
#define L2A  2000   // after pool1
#define TT   1000   // after pool2
#define EPSV 1e-5f

typedef __attribute__((ext_vector_type(2))) float v2f;
typedef __attribute__((ext_vector_type(8))) float v8f;

// ---------------------------------------------------------------------------
// K1/K3: per-channel stats of conv(x)+bias, fused into BN scale/shift.
// grid = C blocks, 256 threads. Recomputes the cheap depthwise conv so the
// full-size conv output never hits HBM.
// ---------------------------------------------------------------------------
__global__ void k_conv_bn_stats(const float* __restrict__ x,
                                const float* __restrict__ w,
                                const float* __restrict__ bias,
                                const float* __restrict__ g,
                                const float* __restrict__ beta,
                                float* __restrict__ scale,
                                float* __restrict__ shift,
                                int Lin) {
    const int c = blockIdx.x;
    const int tid = threadIdx.x;
    const float w0 = w[c * 3 + 0], w1 = w[c * 3 + 1], w2 = w[c * 3 + 2];
    const float bcv = bias[c];
    float s = 0.f, s2 = 0.f;
    for (int b = 0; b < BB; ++b) {
        const float* row = x + ((size_t)(b * CC + c)) * (size_t)Lin;
        for (int l = tid; l < Lin; l += blockDim.x) {
            float xm = (l > 0)       ? row[l - 1] : 0.f;
            float x0 = row[l];
            float xp = (l + 1 < Lin) ? row[l + 1] : 0.f;
            float y = fmaf(w0, xm, fmaf(w1, x0, fmaf(w2, xp, bcv)));
            s += y;
            s2 = fmaf(y, y, s2);
        }
    }
    __shared__ float ss[256];
    __shared__ float sq[256];
    ss[tid] = s; sq[tid] = s2;
    __syncthreads();
    for (int off = 128; off > 0; off >>= 1) {
        if (tid < off) { ss[tid] += ss[tid + off]; sq[tid] += sq[tid + off]; }
        __syncthreads();
    }
    if (tid == 0) {
        float n = (float)BB * (float)Lin;
        float m = ss[0] / n;
        float v = sq[0] / n - m * m;
        float sc = g[c] * rsqrtf(v + EPSV);
        scale[c] = sc;
        shift[c] = beta[c] - m * sc;
    }
}

// ---------------------------------------------------------------------------
// K2/K4: apply conv + BN(scale,shift) + ReLU + maxpool/2.
// One thread per pooled output element; t-fastest mapping => coalesced.
// ---------------------------------------------------------------------------
__global__ void k_conv_bn_relu_pool(const float* __restrict__ in,
                                    const float* __restrict__ w,
                                    const float* __restrict__ bias,
                                    const float* __restrict__ scale,
                                    const float* __restrict__ shift,
                                    float* __restrict__ out,
                                    int Lin) {
    const int lo = Lin >> 1;
    const int idx = blockIdx.x * blockDim.x + threadIdx.x;
    const int total = BB * CC * lo;
    if (idx >= total) return;
    const int l2 = idx % lo;
    const int bc = idx / lo;
    const int c = bc % CC;
    const float w0 = w[c * 3 + 0], w1 = w[c * 3 + 1], w2 = w[c * 3 + 2];
    const float bcv = bias[c];
    const float sc = scale[c], sh = shift[c];
    const float* row = in + (size_t)bc * (size_t)Lin;
    const int l = 2 * l2;
    float xm = (l > 0)       ? row[l - 1] : 0.f;
    float x0 = row[l];
    float x1 = row[l + 1];
    float xp = (l + 2 < Lin) ? row[l + 2] : 0.f;
    float y0 = fmaf(w0, xm, fmaf(w1, x0, fmaf(w2, x1, bcv)));
    float y1 = fmaf(w0, x0, fmaf(w1, x1, fmaf(w2, xp, bcv)));
    float a0 = fmaxf(fmaf(y0, sc, sh), 0.f);
    float a1 = fmaxf(fmaf(y1, sc, sh), 0.f);
    out[(size_t)bc * (size_t)lo + l2] = fmaxf(a0, a1);
}

// ---------------------------------------------------------------------------
// K5: 8192 scalar-hidden LSTMs, one lane per (b,c). Sequential over T=1000.
// Also accumulates sum(h) and sum(h^2) so BN3 stats need no extra pass.
// Per-lane cachelines are re-read for 32 consecutive t => L0-resident.
// ---------------------------------------------------------------------------
__device__ __forceinline__ float sigmf(float x) {
    return 1.f / (1.f + __expf(-x));
}

__global__ void k_lstm(const float* __restrict__ p2,
                       const float* __restrict__ w_ih,
                       const float* __restrict__ w_hh,
                       const float* __restrict__ b_lstm,
                       float* __restrict__ hs,
                       float* __restrict__ hsum_buf,
                       float* __restrict__ hsq_buf,
                       float* __restrict__ ymean) {
    const int tid = blockIdx.x * blockDim.x + threadIdx.x;  // = b*C + c
    if (tid >= BB * CC) return;
    const int c = tid % CC;
    const float4 wi = ((const float4*)w_ih)[c];
    const float4 wh = ((const float4*)w_hh)[c];
    const float4 bl = ((const float4*)b_lstm)[c];
    const size_t base = (size_t)tid * TT;
    float h = 0.f, cst = 0.f, hsum = 0.f, hsq = 0.f;
    for (int t = 0; t < TT; ++t) {
        const float xt = p2[base + t];
        float gi = fmaf(xt, wi.x, fmaf(h, wh.x, bl.x));
        float gf = fmaf(xt, wi.y, fmaf(h, wh.y, bl.y));
        float gg = fmaf(xt, wi.z, fmaf(h, wh.z, bl.z));
        float go = fmaf(xt, wi.w, fmaf(h, wh.w, bl.w));
        float ig = sigmf(gi);
        float fg = sigmf(gf);
        float gv = tanhf(gg);
        float og = sigmf(go);
        cst = fmaf(fg, cst, ig * gv);
        h = og * tanhf(cst);
        hs[base + t] = h;
        hsum += h;
        hsq = fmaf(h, h, hsq);
    }
    hsum_buf[tid] = hsum;
    hsq_buf[tid] = hsq;
    ymean[tid] = hsum * (1.f / (float)TT);
}

// ---------------------------------------------------------------------------
// K6/K7: SE GEMM  Y[m][n] = act( sum_k A[m][k] * W[n][k] )  with A:[64][128],
// W:[128][128].  One wave32 per 16x16 output tile; K-loop of 32
// V_WMMA_F32_16X16X4_F32 ops (full fp32 precision).
// A operand layout: lane l holds A[l%16][2*(l/16)+j]  (j = VGPR 0/1)
// B operand layout: lane l holds B[2*(l/16)+j][l%16] = W[l%16][2*(l/16)+j]
//   => identical row-major float2 load pattern for both operands.
// D layout: VGPR j -> M = j + 8*(l/16), N = l%16.
// ---------------------------------------------------------------------------
__global__ void k_se_gemm(const float* __restrict__ A,
                          const float* __restrict__ W,
                          float* __restrict__ Y,
                          int act) {
    const int lane = threadIdx.x;        // 0..31, EXEC all ones
    const int tm = blockIdx.x >> 3;      // 4 M-tiles (64/16)
    const int tn = blockIdx.x & 7;       // 8 N-tiles (128/16)
    const int half = lane >> 4;
    const int lm = lane & 15;
    const float* arow = A + (size_t)(tm * 16 + lm) * CC;
    const float* wrow = W + (size_t)(tn * 16 + lm) * CC;
    v8f acc = {};
    for (int k0 = 0; k0 < CC; k0 += 4) {
        v2f a = *(const v2f*)(arow + k0 + 2 * half);
        v2f b = *(const v2f*)(wrow + k0 + 2 * half);
        acc = __builtin_amdgcn_wmma_f32_16x16x4_f32(
            /*neg_a=*/false, a, /*neg_b=*/false, b,
            /*c_mod=*/(short)0, acc, /*reuse_a=*/false, /*reuse_b=*/false);
    }
#pragma unroll
    for (int j = 0; j < 8; ++j) {
        const int m = tm * 16 + half * 8 + j;
        const int n = tn * 16 + lm;
        float v = acc[j];
        v = (act == 0) ? fmaxf(v, 0.f) : sigmf(v);
        Y[(size_t)m * CC + n] = v;
    }
}

// ---------------------------------------------------------------------------
// K8: BN3 stats from the analytic identity  z[b,c,t] = hs[b,c,t]*y2[b,c]:
//   sum_z[c]  = sum_b y2[b,c]   * sum_t hs
//   sum_z2[c] = sum_b y2[b,c]^2 * sum_t hs^2
// grid = C blocks of 64 threads (one per b).
// ---------------------------------------------------------------------------
__global__ void k_bn3_stats(const float* __restrict__ y2,
                            const float* __restrict__ hsum_buf,
                            const float* __restrict__ hsq_buf,
                            const float* __restrict__ g3,
                            const float* __restrict__ b3,
                            float* __restrict__ scale,
                            float* __restrict__ shift) {
    const int c = blockIdx.x;
    const int b = threadIdx.x;  // 0..63
    const float yv = y2[b * CC + c];
    float sz = yv * hsum_buf[b * CC + c];
    float szz = yv * yv * hsq_buf[b * CC + c];
    __shared__ float ss[64];
    __shared__ float sq[64];
    ss[b] = sz; sq[b] = szz;
    __syncthreads();
    for (int off = 32; off > 0; off >>= 1) {
        if (b < off) { ss[b] += ss[b + off]; sq[b] += sq[b + off]; }
        __syncthreads();
    }
    if (b == 0) {
        float n = (float)BB * (float)TT;
        float m = ss[0] / n;
        float v = sq[0] / n - m * m;
        float sc = g3[c] * rsqrtf(v + EPSV);
        scale[c] = sc;
        shift[c] = b3[c] - m * sc;
    }
}

// ---------------------------------------------------------------------------
// K9: out = BN3( hs * y2 ).  Coalesced read & write ([B][C][T], t fastest).
// ---------------------------------------------------------------------------
__global__ void k_final(const float* __restrict__ hs,
                        const float* __restrict__ y2,
                        const float* __restrict__ scale,
                        const float* __restrict__ shift,
                        float* __restrict__ out) {
    const int idx = blockIdx.x * blockDim.x + threadIdx.x;
    if (idx >= BB * CC * TT) return;
    const int bc = idx / TT;
    const int c = bc % CC;
    const float z = hs[idx] * y2[bc];
    out[idx] = fmaf(z, scale[c], shift[c]);
}

// ---------------------------------------------------------------------------
// Launcher. Workspace layout (floats), total ~32.81M floats (~132 MB):
//   p1 [B][C][2000] | p2 [B][C][1000] | hs [B][C][1000] |
//   ymean/hsum/hsq/y1/y2 [B*C] each | 6x [C] stat vectors
// ---------------------------------------------------------------------------
extern "C" void kernel_launch(void* const* d_in, const int* in_sizes, int n_in,
                              void* d_out, int out_size, void* d_ws, size_t ws_size,
                              hipStream_t stream) {
    (void)in_sizes; (void)n_in; (void)out_size; (void)ws_size;
    const float* x       = (const float*)d_in[0];
    const float* conv1_w = (const float*)d_in[1];
    const float* conv1_b = (const float*)d_in[2];
    const float* bn1_g   = (const float*)d_in[3];
    const float* bn1_b   = (const float*)d_in[4];
    const float* conv2_w = (const float*)d_in[5];
    const float* conv2_b = (const float*)d_in[6];
    const float* bn2_g   = (const float*)d_in[7];
    const float* bn2_b   = (const float*)d_in[8];
    const float* w_ih    = (const float*)d_in[9];
    const float* w_hh    = (const float*)d_in[10];
    const float* b_lstm  = (const float*)d_in[11];
    const float* se_w1   = (const float*)d_in[12];
    const float* se_w2   = (const float*)d_in[13];
    const float* bn3_g   = (const float*)d_in[14];
    const float* bn3_b   = (const float*)d_in[15];
    float* out = (float*)d_out;

    float* ws = (float*)d_ws;
    size_t off = 0;
    float* p1    = ws + off; off += (size_t)BB * CC * L2A;   // 16,384,000
    float* p2    = ws + off; off += (size_t)BB * CC * TT;    //  8,192,000
    float* hs    = ws + off; off += (size_t)BB * CC * TT;    //  8,192,000
    float* ymean = ws + off; off += (size_t)BB * CC;
    float* hsum  = ws + off; off += (size_t)BB * CC;
    float* hsq   = ws + off; off += (size_t)BB * CC;
    float* y1    = ws + off; off += (size_t)BB * CC;
    float* y2    = ws + off; off += (size_t)BB * CC;
    float* sc1   = ws + off; off += CC;
    float* sh1   = ws + off; off += CC;
    float* sc2   = ws + off; off += CC;
    float* sh2   = ws + off; off += CC;
    float* sc3   = ws + off; off += CC;
    float* sh3   = ws + off; off += CC;

    // Stage 1: conv1 stats -> fused BN params, then conv1+bn1+relu+pool
    k_conv_bn_stats<<<CC, 256, 0, stream>>>(x, conv1_w, conv1_b, bn1_g, bn1_b,
                                            sc1, sh1, LL);
    {
        int total = BB * CC * L2A;
        k_conv_bn_relu_pool<<<(total + 255) / 256, 256, 0, stream>>>(
            x, conv1_w, conv1_b, sc1, sh1, p1, LL);
    }
    // Stage 2: conv2 stats, then conv2+bn2+relu+pool
    k_conv_bn_stats<<<CC, 256, 0, stream>>>(p1, conv2_w, conv2_b, bn2_g, bn2_b,
                                            sc2, sh2, L2A);
    {
        int total = BB * CC * TT;
        k_conv_bn_relu_pool<<<(total + 255) / 256, 256, 0, stream>>>(
            p1, conv2_w, conv2_b, sc2, sh2, p2, L2A);
    }
    // Stage 3: per-channel LSTM scan (+ running sums for SE mean and BN3)
    k_lstm<<<(BB * CC + 255) / 256, 256, 0, stream>>>(p2, w_ih, w_hh, b_lstm,
                                                      hs, hsum, hsq, ymean);
    // Stage 4: SE GEMMs on the matrix pipe (fp32 WMMA 16x16x4)
    k_se_gemm<<<32, 32, 0, stream>>>(ymean, se_w1, y1, /*act=relu*/0);
    k_se_gemm<<<32, 32, 0, stream>>>(y1, se_w2, y2, /*act=sigmoid*/1);
    // Stage 5: BN3 stats (analytic, no pass over hs) + fused scale/shift out
    k_bn3_stats<<<CC, 64, 0, stream>>>(y2, hsum, hsq, bn3_g, bn3_b, sc3, sh3);
    {
        int total = BB * CC * TT;
        k_final<<<(total + 255) / 256, 256, 0, stream>>>(hs, y2, sc3, sh3, out);
    }
}